// Multibox_Loss_90795608637465
// MI455X (gfx1250) — compile-verified
//
#include <hip/hip_runtime.h>
#include <hip/hip_bf16.h>
#include <stdint.h>

typedef __attribute__((ext_vector_type(2))) float        v2f;
typedef __attribute__((ext_vector_type(8))) float        v8f;
typedef __attribute__((ext_vector_type(4))) unsigned int u32x4;
typedef __attribute__((ext_vector_type(4))) int          i32x4;
typedef __attribute__((ext_vector_type(8))) int          i32x8;

#if __has_builtin(__builtin_amdgcn_tensor_load_to_lds) && __has_builtin(__builtin_amdgcn_s_wait_tensorcnt)
#define USE_TDM 1
#else
#define USE_TDM 0
#endif

#if USE_TDM
// Build a Tensor DMA Descriptor (ISA cdna5 §8) for a contiguous 1-row tile of
// `nelem` f32 elements and issue TENSOR_LOAD_TO_LDS.  Tracked by TENSORcnt.
// This toolchain (clang-23 / therock-10.0) exposes the 6-arg builtin:
//   (uint32x4 g0, int32x8 g1, int32x4, int32x4, int32x8, i32 cpol)
// g0/g1 are the ISA D# groups 0/1; remaining descriptor operands are zero
// (= unused groups, <=2D tensor, per VADDR2/3-NULL semantics).
__device__ __forceinline__ void tdm_load_f32_1d(const void* gsrc, void* ldst,
                                                unsigned nelem) {
  unsigned long long ga = (unsigned long long)(uintptr_t)gsrc;
  // Generic pointer to LDS: low 32 bits are the workgroup-relative LDS byte
  // offset (AMDGPU addrspacecast lowering: flat = aperture<<32 | lds_offset).
  unsigned lds = (unsigned)(uintptr_t)ldst;

  u32x4 g0;
  g0[0] = 1u;                                   // count=1 valid, user descriptor
  g0[1] = lds;                                  // lds_addr (bytes)
  g0[2] = (unsigned)(ga & 0xFFFFFFFFu);         // global_addr[31:0]
  g0[3] = (unsigned)((ga >> 32) & 0x01FFFFFFu)  // global_addr[56:32]
          | 0x80000000u;                        // type = 2 ("image")

  i32x8 g1;
  g1[0] = (int)(2u << 16);                      // data_size = 4 bytes; mask/flags 0
  g1[1] = (int)((nelem & 0xFFFFu) << 16);       // tensor_dim0[15:0] @ bits 63:48
  g1[2] = (int)(((nelem >> 16) & 0xFFFFu)       // tensor_dim0[31:16]
          | (1u << 16));                        // tensor_dim1 = 1
  g1[3] = (int)((nelem & 0xFFFFu) << 16);       // tile_dim0 @ bits 127:112
  g1[4] = 1;                                    // tile_dim1 = 1, tile_dim2 = 0
  g1[5] = (int)nelem;                           // tensor_dim0_stride[31:0]
  g1[6] = 0;                                    // stride0 hi / stride1 lo
  g1[7] = 0;

  i32x4 z4 = {0, 0, 0, 0};                      // groups 2/3 unused (<=2D)
  i32x8 z8 = {0, 0, 0, 0, 0, 0, 0, 0};          // extra operand: disabled
  __builtin_amdgcn_tensor_load_to_lds(g0, g1, z4, z4, z8, 0);
}
#endif

// Wave32 all-lane sum.  Uses the matrix pipe: A(16x4)=lane sums (k=1,3 zero),
// B(4x16)=ones -> D[m][n] = acc_m + acc_{m+16} for every n; lanes 0-15 hold
// s_0..s_7 in d[0..7], lanes 16-31 hold s_8..s_15, so one shfl_xor finishes.
// Requires EXEC all ones (call only from fully-active code).
__device__ __forceinline__ float wave_reduce32(float x) {
#if __has_builtin(__builtin_amdgcn_wmma_f32_16x16x4_f32)
  v2f a;  a[0] = x;    a[1] = 0.0f;
  v2f bb; bb[0] = 1.0f; bb[1] = 1.0f;
  v8f c = {};
  v8f d = __builtin_amdgcn_wmma_f32_16x16x4_f32(
      /*neg_a=*/false, a, /*neg_b=*/false, bb,
      /*c_mod=*/(short)0, c, /*reuse_a=*/false, /*reuse_b=*/false);
  float t = ((d[0] + d[1]) + (d[2] + d[3])) + ((d[4] + d[5]) + (d[6] + d[7]));
  t += __shfl_xor(t, 16, 32);
  return t;
#else
  for (int off = 16; off > 0; off >>= 1) x += __shfl_xor(x, off, 32);
  return x;
#endif
}

constexpr int kNT = 128;   // true boxes per batch (fixed by reference)

__global__ __launch_bounds__(256) void mbloss_main(
    const float* __restrict__ pb,   // [B, NP, 4]
    const float* __restrict__ pp,   // [B, NP]
    const float* __restrict__ tb,   // [B, 128, 4]
    const float* __restrict__ tp,   // [B, 128]
    float* __restrict__ partials,   // [gridDim.y * gridDim.x]
    int NP) {
  __shared__ float s_raw_tb[kNT * 4];
  __shared__ float s_raw_tp[kNT];
  __shared__ float s_tx1[kNT], s_ty1[kNT], s_tx2[kNT], s_ty2[kNT];
  __shared__ float s_tp[kNT], s_ta[kNT];
  __shared__ float s_tcx[kNT], s_tcy[kNT], s_twx[kNT], s_twy[kNT];
  __shared__ float s_wred[8];

  const int b   = blockIdx.y;
  const int tid = threadIdx.x;
  const float* tb_b = tb + (size_t)b * kNT * 4;
  const float* tp_b = tp + (size_t)b * kNT;

#if USE_TDM
  if (tid < 32) {                       // wave 0 only issues the DMA
    tdm_load_f32_1d(tb_b, s_raw_tb, kNT * 4);
    tdm_load_f32_1d(tp_b, s_raw_tp, kNT);
    __builtin_amdgcn_s_wait_tensorcnt(0);
  }
#else
  for (int k = tid; k < kNT * 4; k += 256) s_raw_tb[k] = tb_b[k];
  for (int k = tid; k < kNT;     k += 256) s_raw_tp[k] = tp_b[k];
#endif
  __syncthreads();

  if (tid < kNT) {
    float x1 = s_raw_tb[tid * 4 + 0];
    float y1 = s_raw_tb[tid * 4 + 1];
    float x2 = s_raw_tb[tid * 4 + 2];
    float y2 = s_raw_tb[tid * 4 + 3];
    float adx = fabsf(x1 - x2), ady = fabsf(y1 - y2);
    s_tx1[tid] = x1;  s_ty1[tid] = y1;
    s_tx2[tid] = x2;  s_ty2[tid] = y2;
    s_tp[tid]  = s_raw_tp[tid];
    s_ta[tid]  = adx * ady;
    s_tcx[tid] = 150.0f * (x1 + x2);
    s_tcy[tid] = 150.0f * (y1 + y2);
    s_twx[tid] = 300.0f * adx;
    s_twy[tid] = 300.0f * ady;
  }
  __syncthreads();

  float acc = 0.0f;
  const int i = blockIdx.x * 256 + tid;
  if (i < NP) {
    const float4 pv = *(const float4*)(pb + ((size_t)b * NP + i) * 4);
    const float px1 = pv.x, py1 = pv.y, px2 = pv.z, py2 = pv.w;
    const float ppv = pp[(size_t)b * NP + i];
    const float adx = fabsf(px1 - px2), ady = fabsf(py1 - py2);
    const float parea = adx * ady;
    const float pcx = 150.0f * (px1 + px2);
    const float pcy = 150.0f * (py1 + py2);
    const float pwx = 300.0f * adx;
    const float pwy = 300.0f * ady;

#pragma unroll 8
    for (int j = 0; j < kNT; ++j) {
      float xd = fminf(px2, s_tx2[j]) - fmaxf(px1, s_tx1[j]);
      float yd = fminf(py2, s_ty2[j]) - fmaxf(py1, s_ty1[j]);
      float inter = (xd > 0.0f && yd > 0.0f) ? xd * yd : 0.0f;
      // iou < 0.5  <=>  3*inter < area_p + area_t   (inter <= area_p, so the
      // denominator is >= 0; the 0/0 -> NaN case maps to "false" identically).
      bool take_cls = (3.0f * inter) < (parea + s_ta[j]);

      // |pp - tp|^0.1 = exp2(0.1 * log2(|d|));  d==0 -> exp2(-inf) = 0 = pow(0,0.1)
      float dprob = fabsf(ppv - s_tp[j]);
      float cls = __builtin_amdgcn_exp2f(0.1f * __builtin_amdgcn_logf(dprob));

      float dx  = pcx - s_tcx[j];
      float dy  = pcy - s_tcy[j];
      float dwx = pwx - s_twx[j];
      float dwy = pwy - s_twy[j];
      float box = ((dx * dx + dy * dy) + dwx * dwx) + dwy * dwy;

      acc += take_cls ? cls : box;
    }
  }

  // Full-EXEC region: matrix-pipe wave reduction, then LDS across 8 waves.
  float w = wave_reduce32(acc);
  const int wave = tid >> 5, lane = tid & 31;
  if (lane == 0) s_wred[wave] = w;
  __syncthreads();
  if (tid == 0) {
    float s = 0.0f;
#pragma unroll
    for (int k = 0; k < 8; ++k) s += s_wred[k];
    partials[(size_t)blockIdx.y * gridDim.x + blockIdx.x] = s;
  }
}

__global__ __launch_bounds__(256) void mbloss_reduce(
    const float* __restrict__ partials, int n, float* __restrict__ out) {
  __shared__ float s_wred[8];
  float s = 0.0f;
  for (int k = threadIdx.x; k < n; k += 256) s += partials[k];
  s = wave_reduce32(s);
  const int wave = threadIdx.x >> 5, lane = threadIdx.x & 31;
  if (lane == 0) s_wred[wave] = s;
  __syncthreads();
  if (threadIdx.x == 0) {
    float t = 0.0f;
#pragma unroll
    for (int k = 0; k < 8; ++k) t += s_wred[k];
    out[0] = t;
  }
}

extern "C" void kernel_launch(void* const* d_in, const int* in_sizes, int n_in,
                              void* d_out, int out_size, void* d_ws, size_t ws_size,
                              hipStream_t stream) {
  const float* pb = (const float*)d_in[0];  // predict_boxes        [32,8732,4]
  const float* pp = (const float*)d_in[1];  // predict_probabilities[32,8732]
  const float* tb = (const float*)d_in[2];  // true_boxes           [32,128,4]
  const float* tp = (const float*)d_in[3];  // true_probabilities   [32,128]
  float* out = (float*)d_out;

  const int B  = 32;
  const int NP = in_sizes[1] / B;            // 8732
  const int gx = (NP + 255) / 256;           // 35
  float* partials = (float*)d_ws;            // gx*B floats (4.5 KB) of scratch

  dim3 grid(gx, B);
  mbloss_main<<<grid, 256, 0, stream>>>(pb, pp, tb, tp, partials, NP);
  mbloss_reduce<<<1, 256, 0, stream>>>(partials, gx * B, out);
}